// OccupancyToTopology_84937273246103
// MI455X (gfx1250) — compile-verified
//
#include <hip/hip_runtime.h>

typedef __attribute__((ext_vector_type(2))) float v2f;
typedef __attribute__((ext_vector_type(4))) float f32x4;
typedef __attribute__((ext_vector_type(8))) float v8f;

#define ROW   97
#define PLANE (97 * 97)
#define NC    96

__global__ __launch_bounds__(256) void OccupancyToTopology_84937273246103_kernel(
    const float* __restrict__ occ, float* __restrict__ out) {
  const int lane = threadIdx.x & 31;
  const int wave = threadIdx.x >> 5;

  // block -> (x, y) column; wave -> 12 consecutive z cells
  const int x = blockIdx.x / NC;
  const int y = blockIdx.x - x * NC;
  const int z0 = wave * 12;

  const float* base = occ + (x * ROW + y) * ROW;

  // lane-local bit pattern m selects p_k vs (1-p_k); lanes >=16 feed zeros
  const int m = lane & 15;
  const bool b0 = (m & 1) != 0;
  const bool b1 = (m & 2) != 0;
  const bool b2 = (m & 4) != 0;
  const bool b3 = (m & 8) != 0;
  const bool active = lane < 16;

  // z-plane corners: (x,y), (x+1,y), (x+1,y+1), (x,y+1)  [k = 0,1,2,3]
  float q0 = base[z0];
  float q1 = base[z0 + PLANE];
  float q2 = base[z0 + PLANE + ROW];
  float q3 = base[z0 + ROW];

  // D element (lane, r) -> t = 16*(lane&15) + 8*(lane>>4) + r : 8 consecutive t per lane
  const int toff = (m << 4) + ((lane >> 4) << 3);
  float* optr = out + ((size_t)((x * NC + y) * NC + z0)) * 256 + toff;

#pragma unroll
  for (int i = 0; i < 12; ++i) {
    const int z1 = z0 + i + 1;
    float r0 = base[z1];
    float r1 = base[z1 + PLANE];
    float r2 = base[z1 + PLANE + ROW];
    float r3 = base[z1 + ROW];

    // a[m] from z-plane (p0..p3), b[n] from z+1 plane (p4..p7)
    float av = (b0 ? q0 : 1.0f - q0) * (b1 ? q1 : 1.0f - q1) *
               (b2 ? q2 : 1.0f - q2) * (b3 ? q3 : 1.0f - q3);
    float bv = (b0 ? r0 : 1.0f - r0) * (b1 ? r1 : 1.0f - r1) *
               (b2 ? r2 : 1.0f - r2) * (b3 ? r3 : 1.0f - r3);

    // A (16x4): lanes 0-15 vgpr0 = K=0 column; everything else zero.
    // B (4x16): lanes 0-15 vgpr0 = K=0 row; everything else zero.
    v2f A;
    A.x = active ? av : 0.0f;
    A.y = 0.0f;
    v2f B;
    B.x = active ? bv : 0.0f;
    B.y = 0.0f;
    v8f C = {};
    // D = A x B : rank-1 outer product = all 256 topo values for this cell
    v8f D = __builtin_amdgcn_wmma_f32_16x16x4_f32(false, A, false, B, (short)0,
                                                  C, false, false);

    // two b128 stores; wave covers a contiguous 1 KB output row per cell
    f32x4* dst = (f32x4*)optr;
    dst[0] = *(const f32x4*)&D;
    dst[1] = *((const f32x4*)&D + 1);

    // z+1 plane becomes next iteration's z plane
    q0 = r0;
    q1 = r1;
    q2 = r2;
    q3 = r3;
    optr += 256;
  }
}

extern "C" void kernel_launch(void* const* d_in, const int* in_sizes, int n_in,
                              void* d_out, int out_size, void* d_ws,
                              size_t ws_size, hipStream_t stream) {
  (void)in_sizes;
  (void)n_in;
  (void)out_size;
  (void)d_ws;
  (void)ws_size;
  const float* occ = (const float*)d_in[0];
  float* out = (float*)d_out;
  // 96*96 blocks of 256 threads (8 waves); each wave handles 12 z-cells
  OccupancyToTopology_84937273246103_kernel<<<dim3(NC * NC), dim3(256), 0,
                                              stream>>>(occ, out);
}